// NodeModel_35089882808860
// MI455X (gfx1250) — compile-verified
//
#include <hip/hip_runtime.h>

// ---------------------------------------------------------------------------
// MI455X / gfx1250 fused NodeModel:
//   1) edge scatter (sum / max / count) via hardware fp32 global atomics (L2-resident)
//   2) fused  h=[x|sum|max|mean|u[batch]] -> relu(h@W1+b1)@W2+b2 + x
//      using V_WMMA_F32_16X16X4_F32 (full fp32 precision matrix path).
// ---------------------------------------------------------------------------

typedef float v2f __attribute__((ext_vector_type(2)));
typedef float v8f __attribute__((ext_vector_type(8)));

#define NND 100000   // nodes
#define NED 600000   // edges
#define DIM 128      // feature dim
#define HIDN 512     // hidden dim
#define TM  32       // nodes per block (two 16-row WMMA M-tiles)
#define LDH 514      // padded LDS row stride (even -> 8B aligned b64 LDS loads)

// ---------------------------------------------------------------------------
// Kernel 1: initialize accumulators (sum aliased onto d_out), max=-FLT_MAX, cnt=0
// ---------------------------------------------------------------------------
__global__ __launch_bounds__(256) void init_ws_kernel(float* __restrict__ mx,
                                                      float* __restrict__ cnt,
                                                      float* __restrict__ sum)
{
    int i = blockIdx.x * blockDim.x + threadIdx.x;
    if (i < NND * DIM) {
        mx[i]  = -3.402823466e+38f;
        sum[i] = 0.0f;
    }
    if (i < NND) cnt[i] = 0.0f;
}

// ---------------------------------------------------------------------------
// Kernel 2: edge scatter-reduction. One thread per (edge, dim).
// 77M fp32 atomic add + 77M fp32 atomic max; accumulators fit in 192MB L2.
// ---------------------------------------------------------------------------
__global__ __launch_bounds__(256) void edge_agg_kernel(const int*   __restrict__ row,
                                                       const float* __restrict__ ea,
                                                       float* __restrict__ sum,
                                                       float* __restrict__ mx,
                                                       float* __restrict__ cnt)
{
    int i = blockIdx.x * blockDim.x + threadIdx.x;
    if (i >= NED * DIM) return;
    int e = i >> 7;            // edge id
    int d = i & (DIM - 1);     // feature id
    int r = row[e];            // destination segment (source node)
    float v = ea[i];
    unsafeAtomicAdd(&sum[r * DIM + d], v);   // -> global_atomic_add_f32
    unsafeAtomicMax(&mx[r * DIM + d], v);    // -> global_atomic_max_num_f32
    if (d == 0) unsafeAtomicAdd(&cnt[r], 1.0f);
}

// ---------------------------------------------------------------------------
// Kernel 3: fused feature-build + 2-layer MLP with fp32 WMMA.
//   block = 256 threads (8 wave32), 32 nodes per block.
//   GEMM1: 32x513 @ 513x512  (wave w owns hid cols [64w,64w+64), both M-tiles)
//   GEMM2: 32x512 @ 512x128  (wave w owns M-tile w>>2, cols [(w&3)*32, +32))
// WMMA fp32 16x16x4 operand mapping (ISA 7.12.2):
//   A: row = lane&15, K split: lanes 0-15 -> k,k+1 ; lanes 16-31 -> k+2,k+3
//   B: col = lane&15, same K split
//   C/D: col = lane&15, row = vgpr + (lane>=16 ? 8 : 0)
// ---------------------------------------------------------------------------
__global__ __launch_bounds__(256) void mlp_wmma_kernel(
    const float* __restrict__ x,    const float* __restrict__ u,
    const int*   __restrict__ batch,
    const float* __restrict__ W1,   const float* __restrict__ b1,
    const float* __restrict__ W2,   const float* __restrict__ b2,
    const float* __restrict__ sum,  const float* __restrict__ mx,
    const float* __restrict__ cnt,  float* __restrict__ out)
{
    extern __shared__ float h[];       // [TM][LDH]; cols 0..511 = features, reused for hidden
    __shared__ float uval[TM];

    const int tid   = threadIdx.x;
    const int node0 = blockIdx.x * TM;

    // ---- stage: build h = [x | sum | masked-max | mean] (col 512 == u handled in C-init)
    for (int i = tid; i < TM * DIM; i += 256) {
        int m = i >> 7;
        int d = i & (DIM - 1);
        int n = node0 + m;
        float xv = x  [n * DIM + d];
        float sv = sum[n * DIM + d];
        float mv = mx [n * DIM + d];
        float cv = cnt[n];
        h[m * LDH + d]            = xv;
        h[m * LDH + DIM + d]      = sv;
        h[m * LDH + 2 * DIM + d]  = (cv > 0.0f) ? mv : 0.0f;
        h[m * LDH + 3 * DIM + d]  = sv / fmaxf(cv, 1.0f);
    }
    if (tid < TM) uval[tid] = u[batch[node0 + tid]];
    __syncthreads();

    const int wave  = tid >> 5;
    const int lane  = tid & 31;
    const int lhalf = lane >> 4;       // 0: lanes 0-15, 1: lanes 16-31
    const int lcol  = lane & 15;

    // ---- GEMM1: acc init = b1[col] + u[batch[n]] * W1[512][col]  (the k=512 column)
    v8f acc[2][4];
    {
        const float* w1last = W1 + 512 * HIDN;
#pragma unroll
        for (int mt = 0; mt < 2; ++mt)
#pragma unroll
            for (int t = 0; t < 4; ++t) {
                int col  = wave * 64 + t * 16 + lcol;
                float bb = b1[col];
                float wl = w1last[col];
#pragma unroll
                for (int v = 0; v < 8; ++v)
                    acc[mt][t][v] = bb + uval[mt * 16 + v + lhalf * 8] * wl;
            }
    }

    for (int k = 0; k < HIDN; k += 4) {
        int ka = k + lhalf * 2;
        v2f a0, a1;
        a0.x = h[lcol * LDH + ka];            a0.y = h[lcol * LDH + ka + 1];
        a1.x = h[(16 + lcol) * LDH + ka];     a1.y = h[(16 + lcol) * LDH + ka + 1];
#pragma unroll
        for (int t = 0; t < 4; ++t) {
            int col = wave * 64 + t * 16 + lcol;
            v2f b;
            b.x = W1[ka * HIDN + col];
            b.y = W1[(ka + 1) * HIDN + col];
            acc[0][t] = __builtin_amdgcn_wmma_f32_16x16x4_f32(
                false, a0, false, b, (short)0, acc[0][t], false, false);
            acc[1][t] = __builtin_amdgcn_wmma_f32_16x16x4_f32(
                false, a1, false, b, (short)0, acc[1][t], false, false);
        }
    }

    __syncthreads();                     // all waves finished reading h
    // ReLU + write hidden activations back into the same LDS tile
#pragma unroll
    for (int mt = 0; mt < 2; ++mt)
#pragma unroll
        for (int t = 0; t < 4; ++t) {
            int col = wave * 64 + t * 16 + lcol;
#pragma unroll
            for (int v = 0; v < 8; ++v)
                h[(mt * 16 + v + lhalf * 8) * LDH + col] = fmaxf(acc[mt][t][v], 0.0f);
        }
    __syncthreads();

    // ---- GEMM2: acc init = b2[col] + x[n][col] (residual folded in)
    const int mt2   = wave >> 2;
    const int cbase = (wave & 3) * 32;
    v8f o[2];
#pragma unroll
    for (int t = 0; t < 2; ++t) {
        int col  = cbase + t * 16 + lcol;
        float bb = b2[col];
#pragma unroll
        for (int v = 0; v < 8; ++v) {
            int n = node0 + mt2 * 16 + v + lhalf * 8;
            o[t][v] = bb + x[n * DIM + col];
        }
    }

    for (int k = 0; k < HIDN; k += 4) {
        int ka = k + lhalf * 2;
        v2f a;
        a.x = h[(mt2 * 16 + lcol) * LDH + ka];
        a.y = h[(mt2 * 16 + lcol) * LDH + ka + 1];
#pragma unroll
        for (int t = 0; t < 2; ++t) {
            int col = cbase + t * 16 + lcol;
            v2f b;
            b.x = W2[ka * DIM + col];
            b.y = W2[(ka + 1) * DIM + col];
            o[t] = __builtin_amdgcn_wmma_f32_16x16x4_f32(
                false, a, false, b, (short)0, o[t], false, false);
        }
    }

#pragma unroll
    for (int t = 0; t < 2; ++t) {
        int col = cbase + t * 16 + lcol;
#pragma unroll
        for (int v = 0; v < 8; ++v) {
            int n = node0 + mt2 * 16 + v + lhalf * 8;
            out[n * DIM + col] = o[t][v];
        }
    }
}

// ---------------------------------------------------------------------------
extern "C" void kernel_launch(void* const* d_in, const int* in_sizes, int n_in,
                              void* d_out, int out_size, void* d_ws, size_t ws_size,
                              hipStream_t stream)
{
    const float* x          = (const float*)d_in[0];
    const int*   edge_index = (const int*)  d_in[1];   // [2][E]; row = edge_index[0]
    const float* edge_attr  = (const float*)d_in[2];
    const float* u          = (const float*)d_in[3];
    const int*   batch      = (const int*)  d_in[4];
    const float* W1         = (const float*)d_in[5];   // [513][512]
    const float* b1         = (const float*)d_in[6];
    const float* W2         = (const float*)d_in[7];   // [512][128]
    const float* b2         = (const float*)d_in[8];
    float*       out        = (float*)d_out;

    // Workspace: sum is aliased onto d_out (each block reads its sum rows into
    // LDS before writing the same out rows). ws holds max [N*D] + cnt [N].
    float* sum = out;
    float* mx  = (float*)d_ws;
    float* cnt = mx + (size_t)NND * DIM;

    const int* row = edge_index;   // first E entries = source node ids

    {
        int total  = NND * DIM;
        int blocks = (total + 255) / 256;
        init_ws_kernel<<<blocks, 256, 0, stream>>>(mx, cnt, sum);
    }
    {
        int total  = NED * DIM;
        int blocks = (total + 255) / 256;
        edge_agg_kernel<<<blocks, 256, 0, stream>>>(row, edge_attr, sum, mx, cnt);
    }
    {
        size_t shmem = (size_t)TM * LDH * sizeof(float);   // ~66 KB dynamic LDS
        mlp_wmma_kernel<<<NND / TM, 256, shmem, stream>>>(
            x, u, batch, W1, b1, W2, b2, sum, mx, cnt, out);
    }
}